// TATransEModel_43456479101627
// MI455X (gfx1250) — compile-verified
//
#include <hip/hip_runtime.h>
#include <hip/hip_bf16.h>
#include <math.h>

typedef __attribute__((ext_vector_type(2))) float v2f;
typedef __attribute__((ext_vector_type(8))) float v8f;

#define DHID 128
#define GATES 512
#define EPSV 1e-4f

#if defined(__has_builtin)
#if __has_builtin(__builtin_amdgcn_global_load_async_to_lds_b128)
#define HAVE_ASYNC_LDS 1
#endif
#endif

// Pointer/vector types matching the probed builtin signature:
// param1 = addrspace(1) int4* (HIP prints AS1 as "__device__ *")
typedef int v4i_ __attribute__((vector_size(16)));
typedef __attribute__((address_space(1))) v4i_* gptr_v4i;
typedef __attribute__((address_space(3))) v4i_* lptr_v4i;

__device__ __forceinline__ v8f wmma_f32_k4(v2f a, v2f b, v8f c) {
  // 8 args: (neg_a, A, neg_b, B, c_mod, C, reuse_a, reuse_b)
  return __builtin_amdgcn_wmma_f32_16x16x4_f32(false, a, false, b, (short)0, c, false, false);
}

__device__ __forceinline__ float sigmoidf_(float x) {
  return 1.0f / (1.0f + __expf(-x));
}

__device__ __forceinline__ void wait_asynccnt0() {
#if defined(__has_builtin)
#if __has_builtin(__builtin_amdgcn_s_wait_asynccnt)
  __builtin_amdgcn_s_wait_asynccnt(0);
  return;
#else
  asm volatile("s_wait_asynccnt 0x0" ::: "memory");
  return;
#endif
#else
  asm volatile("s_wait_asynccnt 0x0" ::: "memory");
#endif
}

// ---------------------------------------------------------------------------
// Kernel A: per-entity squared norms (one wave per row, float4 loads)
// ---------------------------------------------------------------------------
__global__ __launch_bounds__(256) void ent_sq_kernel(const float* __restrict__ ent,
                                                     float* __restrict__ e2, int E) {
  int row  = blockIdx.x * 8 + (threadIdx.x >> 5);
  int lane = threadIdx.x & 31;
  if (row >= E) return;
  const float4* p = (const float4*)(ent + (size_t)row * DHID);
  float4 v = p[lane];                       // 32 lanes x float4 = 128 floats
  float s = v.x * v.x + v.y * v.y + v.z * v.z + v.w * v.w;
  #pragma unroll
  for (int off = 16; off > 0; off >>= 1) s += __shfl_xor(s, off, 32);
  if (lane == 0) e2[row] = s;
}

// ---------------------------------------------------------------------------
// Kernel B: LSTM over 9-token sequence, 16 batch rows per block, 8 waves.
// gates(16,512) = x@w_ih^T + h@w_hh^T + bias via V_WMMA_F32_16X16X4_F32.
// ---------------------------------------------------------------------------
__global__ __launch_bounds__(256) void lstm_kernel(
    const int* __restrict__ pos_r, const int* __restrict__ pos_tem,
    const float* __restrict__ rel_emb, const float* __restrict__ tem_emb,
    const float* __restrict__ w_ih, const float* __restrict__ w_hh,
    const float* __restrict__ b_ih, const float* __restrict__ b_hh,
    float* __restrict__ rseq, int T) {
  __shared__ float xbuf[16][DHID];
  __shared__ float hbuf[16][DHID];
  __shared__ float cbuf[16][DHID];
  __shared__ float gbuf[16][GATES];

  const int tid   = threadIdx.x;
  const int wave  = tid >> 5;
  const int lane  = tid & 31;
  const int lhalf = lane >> 4;   // 0: K pair {0,1}; 1: K pair {2,3}
  const int lmod  = lane & 15;
  const int r0    = blockIdx.x * 16;

  for (int e = tid; e < 16 * DHID; e += 256) {
    ((float*)hbuf)[e] = 0.0f;
    ((float*)cbuf)[e] = 0.0f;
  }
  __syncthreads();

  for (int t = 0; t <= T; ++t) {
    // stage x_t: token 0 = rel embedding, tokens 1..T = temporal embeddings
    for (int e = tid; e < 16 * DHID; e += 256) {
      int m = e >> 7, k = e & (DHID - 1);
      int b = r0 + m;
      const float* src = (t == 0) ? (rel_emb + (size_t)pos_r[b] * DHID)
                                  : (tem_emb + (size_t)pos_tem[b * T + (t - 1)] * DHID);
      ((float*)xbuf)[e] = src[k];
    }
    __syncthreads();

    v8f acc[4];
    v8f zero = {};
    #pragma unroll
    for (int n = 0; n < 4; ++n) acc[n] = zero;

    for (int kb = 0; kb < 32; ++kb) {
      int k = kb * 4 + lhalf * 2;
      v2f ax, ah;
      ax.x = xbuf[lmod][k]; ax.y = xbuf[lmod][k + 1];
      ah.x = hbuf[lmod][k]; ah.y = hbuf[lmod][k + 1];
      #pragma unroll
      for (int n = 0; n < 4; ++n) {
        int col = wave * 64 + n * 16 + lmod;    // gate column (0..511)
        const float* wr1 = w_ih + (size_t)col * DHID + k;
        const float* wr2 = w_hh + (size_t)col * DHID + k;
        v2f bx, bh;
        bx.x = wr1[0]; bx.y = wr1[1];
        bh.x = wr2[0]; bh.y = wr2[1];
        acc[n] = wmma_f32_k4(ax, bx, acc[n]);
        acc[n] = wmma_f32_k4(ah, bh, acc[n]);
      }
    }

    // bias + scatter gates to LDS
    #pragma unroll
    for (int n = 0; n < 4; ++n) {
      int col = wave * 64 + n * 16 + lmod;
      float bias = b_ih[col] + b_hh[col];
      #pragma unroll
      for (int g = 0; g < 8; ++g) {
        int m = g + lhalf * 8;                  // C/D layout: M = g + 8*(lane>=16)
        gbuf[m][col] = acc[n][g] + bias;
      }
    }
    __syncthreads();

    // elementwise cell update: i,f,g,o split along gate columns
    for (int e = tid; e < 16 * DHID; e += 256) {
      int m = e >> 7, j = e & (DHID - 1);
      float ig = sigmoidf_(gbuf[m][j]);
      float fg = sigmoidf_(gbuf[m][DHID + j]);
      float gg = tanhf(gbuf[m][2 * DHID + j]);
      float og = sigmoidf_(gbuf[m][3 * DHID + j]);
      float c  = fg * cbuf[m][j] + ig * gg;
      cbuf[m][j] = c;
      hbuf[m][j] = og * tanhf(c);
    }
    __syncthreads();
  }

  for (int e = tid; e < 16 * DHID; e += 256) {
    int m = e >> 7, j = e & (DHID - 1);
    rseq[(size_t)(r0 + m) * DHID + j] = hbuf[m][j];
  }
}

// ---------------------------------------------------------------------------
// Kernel C: q1 = h_e + rseq, q2 = t_e - rseq, plus row squared norms.
// ---------------------------------------------------------------------------
__global__ __launch_bounds__(256) void qprep_kernel(
    const int* __restrict__ pos_h, const int* __restrict__ pos_t,
    const float* __restrict__ ent, const float* __restrict__ rseq,
    float* __restrict__ q1, float* __restrict__ q2,
    float* __restrict__ q1sq, float* __restrict__ q2sq, int B) {
  int row  = blockIdx.x * 8 + (threadIdx.x >> 5);
  int lane = threadIdx.x & 31;
  if (row >= B) return;
  const float4* he = (const float4*)(ent + (size_t)pos_h[row] * DHID);
  const float4* te = (const float4*)(ent + (size_t)pos_t[row] * DHID);
  const float4* rs = (const float4*)(rseq + (size_t)row * DHID);
  float4 h4 = he[lane], t4 = te[lane], r4 = rs[lane];
  float4 a, b;
  a.x = h4.x + r4.x; a.y = h4.y + r4.y; a.z = h4.z + r4.z; a.w = h4.w + r4.w;
  b.x = t4.x - r4.x; b.y = t4.y - r4.y; b.z = t4.z - r4.z; b.w = t4.w - r4.w;
  ((float4*)(q1 + (size_t)row * DHID))[lane] = a;
  ((float4*)(q2 + (size_t)row * DHID))[lane] = b;
  float s1 = a.x * a.x + a.y * a.y + a.z * a.z + a.w * a.w;
  float s2 = b.x * b.x + b.y * b.y + b.z * b.z + b.w * b.w;
  #pragma unroll
  for (int off = 16; off > 0; off >>= 1) {
    s1 += __shfl_xor(s1, off, 32);
    s2 += __shfl_xor(s2, off, 32);
  }
  if (lane == 0) { q1sq[row] = s1; q2sq[row] = s2; }
}

// ---------------------------------------------------------------------------
// Kernel D: fused distances + column softmax + write.
// Block = 8 waves = all 512 batch rows for one 16-entity strip; z stays in
// WMMA accumulators; softmax over batch axis reduced via shfl_xor + LDS.
// Ent tile (8KB) staged to LDS via async Global->LDS (ASYNCcnt) when the
// toolchain exposes the builtin; q row norms staged to LDS as well.
// ---------------------------------------------------------------------------
__global__ __launch_bounds__(256) void dist_softmax_kernel(
    const float* __restrict__ q1, const float* __restrict__ q2,
    const float* __restrict__ q1sq, const float* __restrict__ q2sq,
    const float* __restrict__ ent, const float* __restrict__ e2,
    float* __restrict__ out, int E, int B) {
  __shared__ float ebuf[16][DHID];   // 8 KB entity tile (16 rows x 128)
  __shared__ float sqbuf[512];       // q row squared norms
  __shared__ float red[8][16];

  const int tid   = threadIdx.x;
  const int wave  = tid >> 5;
  const int lane  = tid & 31;
  const int lhalf = lane >> 4;
  const int lmod  = lane & 15;
  const int col   = blockIdx.x * 16 + lmod;     // entity column
  const float* q   = blockIdx.y ? q2 : q1;
  const float* qsq = blockIdx.y ? q2sq : q1sq;

  // --- stage entity tile (contiguous 16*128 floats) into LDS ---
  const float* gbase = ent + (size_t)blockIdx.x * 16 * DHID;
#if defined(HAVE_ASYNC_LDS)
  for (int c = tid; c < 512; c += 256) {       // 512 x 16B chunks
    __builtin_amdgcn_global_load_async_to_lds_b128(
        (gptr_v4i)(float*)(gbase + c * 4),
        (lptr_v4i)(&((float*)ebuf)[c * 4]),
        0, 0);
  }
#else
  for (int c = tid; c < 512; c += 256) {
    ((float4*)ebuf)[c] = ((const float4*)gbase)[c];
  }
#endif
  // --- stage q row norms ---
  for (int i = tid; i < B; i += 256) sqbuf[i] = qsq[i];
#if defined(HAVE_ASYNC_LDS)
  wait_asynccnt0();
#endif
  __syncthreads();

  v8f acc[4];
  v8f zero = {};
  #pragma unroll
  for (int n = 0; n < 4; ++n) acc[n] = zero;

  for (int kb = 0; kb < 32; ++kb) {
    int k = kb * 4 + lhalf * 2;
    v2f b;
    b.x = ebuf[lmod][k];
    b.y = ebuf[lmod][k + 1];
    #pragma unroll
    for (int mt = 0; mt < 4; ++mt) {
      int arow = wave * 64 + mt * 16 + lmod;
      const float* qr = q + (size_t)arow * DHID + k;
      v2f a; a.x = qr[0]; a.y = qr[1];
      acc[mt] = wmma_f32_k4(a, b, acc[mt]);
    }
  }

  // z = 1 / (|q|^2 - 2 q.e + |e|^2 + eps), in place; track column max
  float e2c  = e2[col];
  float zmax = -1e30f;
  #pragma unroll
  for (int mt = 0; mt < 4; ++mt) {
    #pragma unroll
    for (int g = 0; g < 8; ++g) {
      int row = wave * 64 + mt * 16 + g + lhalf * 8;
      float d = sqbuf[row] - 2.0f * acc[mt][g] + e2c;
      float z = 1.0f / (d + EPSV);
      acc[mt][g] = z;
      zmax = fmaxf(zmax, z);
    }
  }
  zmax = fmaxf(zmax, __shfl_xor(zmax, 16, 32));   // merge lane l <-> l+16 (same column)
  if (lane < 16) red[wave][lane] = zmax;
  __syncthreads();
  float cmax = red[0][lmod];
  #pragma unroll
  for (int w2 = 1; w2 < 8; ++w2) cmax = fmaxf(cmax, red[w2][lmod]);
  __syncthreads();

  float sum = 0.0f;
  #pragma unroll
  for (int mt = 0; mt < 4; ++mt) {
    #pragma unroll
    for (int g = 0; g < 8; ++g) {
      float ev = __expf(acc[mt][g] - cmax);
      acc[mt][g] = ev;
      sum += ev;
    }
  }
  sum += __shfl_xor(sum, 16, 32);
  if (lane < 16) red[wave][lane] = sum;
  __syncthreads();
  float csum = 0.0f;
  #pragma unroll
  for (int w2 = 0; w2 < 8; ++w2) csum += red[w2][lmod];
  float inv = 1.0f / csum;

  size_t rowoff = (size_t)blockIdx.y * (size_t)B;
  #pragma unroll
  for (int mt = 0; mt < 4; ++mt) {
    #pragma unroll
    for (int g = 0; g < 8; ++g) {
      int row = wave * 64 + mt * 16 + g + lhalf * 8;
      out[(rowoff + (size_t)row) * (size_t)E + col] = acc[mt][g] * inv;
    }
  }
}

// ---------------------------------------------------------------------------
// Kernel E: targets = concat(pos_t, pos_h) appended after pred.
// ---------------------------------------------------------------------------
__global__ void target_kernel(const int* __restrict__ pos_t, const int* __restrict__ pos_h,
                              float* __restrict__ out, int B, size_t base) {
  int i = blockIdx.x * 256 + threadIdx.x;
  if (i < B) {
    out[base + i]     = (float)pos_t[i];
    out[base + B + i] = (float)pos_h[i];
  }
}

extern "C" void kernel_launch(void* const* d_in, const int* in_sizes, int n_in,
                              void* d_out, int out_size, void* d_ws, size_t ws_size,
                              hipStream_t stream) {
  const int*   pos_h   = (const int*)d_in[2];
  const int*   pos_t   = (const int*)d_in[3];
  const int*   pos_r   = (const int*)d_in[4];
  const int*   pos_tem = (const int*)d_in[5];
  const float* ent_emb = (const float*)d_in[10];
  const float* rel_emb = (const float*)d_in[11];
  const float* tem_emb = (const float*)d_in[12];
  const float* w_ih    = (const float*)d_in[13];
  const float* w_hh    = (const float*)d_in[14];
  const float* b_ih    = (const float*)d_in[15];
  const float* b_hh    = (const float*)d_in[16];

  const int B = in_sizes[2];
  const int T = in_sizes[5] / B;
  const int E = in_sizes[10] / DHID;

  float* out  = (float*)d_out;
  float* ws   = (float*)d_ws;
  float* rseq = ws;
  float* q1   = rseq + (size_t)B * DHID;
  float* q2   = q1   + (size_t)B * DHID;
  float* q1sq = q2   + (size_t)B * DHID;
  float* q2sq = q1sq + B;
  float* e2   = q2sq + B;                     // E floats; total < 1 MB

  ent_sq_kernel<<<dim3((E + 7) / 8), 256, 0, stream>>>(ent_emb, e2, E);
  lstm_kernel<<<dim3(B / 16), 256, 0, stream>>>(pos_r, pos_tem, rel_emb, tem_emb,
                                                w_ih, w_hh, b_ih, b_hh, rseq, T);
  qprep_kernel<<<dim3((B + 7) / 8), 256, 0, stream>>>(pos_h, pos_t, ent_emb, rseq,
                                                      q1, q2, q1sq, q2sq, B);
  dist_softmax_kernel<<<dim3(E / 16, 2), 256, 0, stream>>>(q1, q2, q1sq, q2sq,
                                                           ent_emb, e2, out, E, B);
  target_kernel<<<dim3((B + 255) / 256), 256, 0, stream>>>(pos_t, pos_h, out, B,
                                                           (size_t)2 * (size_t)B * (size_t)E);
}